// TimeSpaceStrainModel_86397562126999
// MI455X (gfx1250) — compile-verified
//
#include <hip/hip_runtime.h>
#include <math.h>

// Problem dims (fixed by the reference)
#define T_DIM 100
#define R_DIM 300
#define S_DIM 500
#define P_DIM 8
#define RC_DIM 50

// Padded GEMM dims
#define MP 304            // rows: 19 * 16
#define NP 512            // cols: 8 * 64 (4 tiles of 16 per wave)
#define KP1 150           // K pairs for GEMM1 (K = R = 300)
#define KP2 250           // K pairs for GEMM2 (K = S = 500)

constexpr float DEATH_RATE = 0.02f;
constexpr float LOG2PI_F   = 1.8378770664093453f;

typedef __attribute__((ext_vector_type(2))) float v2f;
typedef __attribute__((ext_vector_type(8))) float v8f;

// ---------------------------------------------------------------------------
// small helpers
// ---------------------------------------------------------------------------
__device__ inline float negbin_lp(float k, float rate, float od) {
  // OverdispersedPoisson -> NB(r, p), q = 1/(1+od*rate)
  float q = 1.0f / (1.0f + od * rate);
  float p = 1.0f - q;
  float r = rate * q / p;
  return lgammaf(k + r) - lgammaf(r) - lgammaf(k + 1.0f)
       + r * logf(q) + k * logf(p);
}

__device__ inline float wave32_sum(float v) {
  #pragma unroll
  for (int off = 16; off >= 1; off >>= 1) v += __shfl_xor(v, off, 32);
  return v;
}

// ---------------------------------------------------------------------------
// init / prep kernels (write the zero pads every call -> deterministic ws)
// ---------------------------------------------------------------------------
__global__ void init_out_k(float* out) { out[0] = 0.0f; }

// transitT[m, k] = sum_p transit_data[k, m, p] * transit_rate[p], m padded to MP
__global__ void build_transitT_k(const float* __restrict__ td,
                                 const float* __restrict__ rate,
                                 float* __restrict__ transitT) {
  int idx = blockIdx.x * blockDim.x + threadIdx.x;   // idx = m*R + k
  if (idx >= MP * R_DIM) return;
  int m = idx / R_DIM, k = idx % R_DIM;
  float acc = 0.0f;
  if (m < R_DIM) {
    #pragma unroll
    for (int p = 0; p < P_DIM; ++p)
      acc += td[((size_t)k * R_DIM + m) * P_DIM + p] * rate[p];
  }
  transitT[idx] = acc;
}

// mutP[kp][n][i] = eye + mr*mm[2kp+i, n], n padded to NP (zeros)
__global__ void build_mutP_k(const float* __restrict__ mm,
                             const float* __restrict__ mrate,
                             float* __restrict__ mutP) {
  int idx = blockIdx.x * blockDim.x + threadIdx.x;   // idx = kp*NP + n
  if (idx >= KP2 * NP) return;
  int kp = idx / NP, n = idx % NP;
  v2f v = {};
  if (n < S_DIM) {
    int k0 = 2 * kp, k1 = 2 * kp + 1;
    float mr = mrate[0];
    v.x = ((k0 == n) ? 1.0f : 0.0f) + mr * mm[(size_t)k0 * S_DIM + n];
    v.y = ((k1 == n) ? 1.0f : 0.0f) + mr * mm[(size_t)k1 * S_DIM + n];
  }
  ((v2f*)mutP)[idx] = v;
}

// tmpP[t][kp][n][i] = infections[t, 2kp+i, n] * R0 * Rs[n] * Rtr[t, 2kp+i]
// n padded to NP (zeros)
__global__ void build_tmpP_k(const float* __restrict__ inf,
                             const float* __restrict__ Rtr,
                             const float* __restrict__ Rs,
                             const float* __restrict__ R0,
                             float* __restrict__ tmpP) {
  int idx = blockIdx.x * blockDim.x + threadIdx.x;   // idx = (t*KP1 + kp)*NP + n
  const int N = (T_DIM - 1) * KP1 * NP;
  if (idx >= N) return;
  int n  = idx % NP;
  int kp = (idx / NP) % KP1;
  int t  = idx / (KP1 * NP);
  v2f v = {};
  if (n < S_DIM) {
    int r0 = 2 * kp, r1 = 2 * kp + 1;
    float sc = R0[0] * Rs[n];
    v.x = inf[((size_t)t * R_DIM + r0) * S_DIM + n] * sc * Rtr[t * R_DIM + r0];
    v.y = inf[((size_t)t * R_DIM + r1) * S_DIM + n] * sc * Rtr[t * R_DIM + r1];
  }
  ((v2f*)tmpP)[idx] = v;
}

// ---------------------------------------------------------------------------
// GEMM1:  pred1[t, m, n] = sum_k transitT[m, k] * tmp[t, k, n]
//   f32 WMMA 16x16x4; each wave owns a 16x64 strip (4 accumulators share A).
//   Fully padded operands -> zero masks, zero clamps, unconditional stores.
// ---------------------------------------------------------------------------
__global__ __launch_bounds__(128)
void gemm1_wmma_k(const float* __restrict__ transitT,
                  const float* __restrict__ tmpP,
                  float* __restrict__ pred1) {
  const int lane = threadIdx.x & 31;
  const int wave = threadIdx.x >> 5;
  const int half = lane >> 4;               // K-pair selector per half-wave
  const int lrow = lane & 15;
  const int m0 = blockIdx.x * 16;
  const int n0 = (blockIdx.y * 4 + wave) * 64;
  const int t  = blockIdx.z;

  const v2f* __restrict__ Arow = (const v2f*)(transitT + (size_t)(m0 + lrow) * R_DIM);
  const v2f* __restrict__ Bp   = (const v2f*)(tmpP + (size_t)t * KP1 * NP * 2);
  const int col0 = n0 + lrow;

  v8f c0 = {}, c1 = {}, c2 = {}, c3 = {};
  for (int k2 = 0; k2 < KP1; k2 += 2) {     // 75 iterations, K = 300
    const int kp = k2 + half;
    v2f a  = Arow[kp];
    const v2f* brow = Bp + (size_t)kp * NP + col0;
    v2f b0 = brow[0];
    v2f b1 = brow[16];
    v2f b2 = brow[32];
    v2f b3 = brow[48];
    c0 = __builtin_amdgcn_wmma_f32_16x16x4_f32(false, a, false, b0, (short)0, c0, false, false);
    c1 = __builtin_amdgcn_wmma_f32_16x16x4_f32(false, a, false, b1, (short)0, c1, false, false);
    c2 = __builtin_amdgcn_wmma_f32_16x16x4_f32(false, a, false, b2, (short)0, c2, false, false);
    c3 = __builtin_amdgcn_wmma_f32_16x16x4_f32(false, a, false, b3, (short)0, c3, false, false);
  }

  float* __restrict__ P = pred1 + (size_t)t * MP * NP;
  #pragma unroll
  for (int j = 0; j < 8; ++j) {
    const int row = m0 + j + 8 * half;      // documented 16x16 f32 C/D layout
    float* pr = P + (size_t)row * NP + col0;
    pr[0]  = c0[j];
    pr[16] = c1[j];
    pr[32] = c2[j];
    pr[48] = c3[j];
  }
}

// ---------------------------------------------------------------------------
// GEMM2 + fused log-normal epilogue:
//   pred2[t, m, n] = sum_k pred1[t, m, k] * mut[k, n]  (stays in registers)
//   lp = lognormal_lp(infections[t+1], log(pred)-s2/2, sqrt(s2)),
//   s2 = log1p(1/pred + od); branchless validity-masked accumulation.
// ---------------------------------------------------------------------------
__global__ __launch_bounds__(128)
void gemm2_wmma_ll_k(const float* __restrict__ pred1,
                     const float* __restrict__ mutP,
                     const float* __restrict__ inf,
                     const float* __restrict__ od_p,
                     float* __restrict__ out) {
  const int lane = threadIdx.x & 31;
  const int wave = threadIdx.x >> 5;
  const int half = lane >> 4;
  const int lrow = lane & 15;
  const int m0 = blockIdx.x * 16;
  const int n0 = (blockIdx.y * 4 + wave) * 64;
  const int t  = blockIdx.z;

  const v2f* __restrict__ Arow =
      (const v2f*)(pred1 + (size_t)t * MP * NP + (size_t)(m0 + lrow) * NP);
  const v2f* __restrict__ Bp = (const v2f*)mutP;
  const int col0 = n0 + lrow;

  v8f c0 = {}, c1 = {}, c2 = {}, c3 = {};
  for (int k2 = 0; k2 < KP2; k2 += 2) {     // 125 iterations, K = 500
    const int kp = k2 + half;
    v2f a  = Arow[kp];
    const v2f* brow = Bp + (size_t)kp * NP + col0;
    v2f b0 = brow[0];
    v2f b1 = brow[16];
    v2f b2 = brow[32];
    v2f b3 = brow[48];
    c0 = __builtin_amdgcn_wmma_f32_16x16x4_f32(false, a, false, b0, (short)0, c0, false, false);
    c1 = __builtin_amdgcn_wmma_f32_16x16x4_f32(false, a, false, b1, (short)0, c1, false, false);
    c2 = __builtin_amdgcn_wmma_f32_16x16x4_f32(false, a, false, b2, (short)0, c2, false, false);
    c3 = __builtin_amdgcn_wmma_f32_16x16x4_f32(false, a, false, b3, (short)0, c3, false, false);
  }

  // fused epilogue: RelaxedPoisson / log-normal lp vs infections[t+1]
  const float od = od_p[0];
  const float* __restrict__ I1 = inf + (size_t)(t + 1) * R_DIM * S_DIM;
  float acc = 0.0f;
  #pragma unroll
  for (int j = 0; j < 8; ++j) {
    const int row  = m0 + j + 8 * half;
    const int rowc = row < R_DIM ? row : R_DIM - 1;
    const float rm = row < R_DIM ? 1.0f : 0.0f;
    const float* ir = I1 + (size_t)rowc * S_DIM;
    #pragma unroll
    for (int q = 0; q < 4; ++q) {
      const int col  = col0 + 16 * q;
      const int colc = col < S_DIM ? col : S_DIM - 1;
      const float vm = rm * (col < S_DIM ? 1.0f : 0.0f);
      float cv = (q == 0) ? c0[j] : (q == 1) ? c1[j] : (q == 2) ? c2[j] : c3[j];
      float pred = fmaxf(cv, 1e-3f);            // clamp (straight-through)
      float s2 = log1pf(1.0f / pred + od);
      float m  = logf(pred) - 0.5f * s2;
      float lx = logf(ir[colc]);                // clamped read: always finite
      float d  = lx - m;
      acc += vm * (-lx - 0.5f * logf(s2) - 0.5f * LOG2PI_F - d * d / (2.0f * s2));
    }
  }
  acc = wave32_sum(acc);
  if (lane == 0) atomicAdd(out, acc);
}

// ---------------------------------------------------------------------------
// Rtr drift:  lp = lognormal(Rtr[t+1]/Rtr[t]; 0, scale), summed.
// ---------------------------------------------------------------------------
__global__ void drift_ll_k(const float* __restrict__ Rtr,
                           const float* __restrict__ scale_p,
                           float* __restrict__ out) {
  __shared__ float sred[256];
  const int N = (T_DIM - 1) * R_DIM;
  const float sc = scale_p[0];
  float acc = 0.0f;
  for (int i = blockIdx.x * blockDim.x + threadIdx.x; i < N;
       i += gridDim.x * blockDim.x) {
    int t = i / R_DIM, r = i % R_DIM;
    float x  = Rtr[(t + 1) * R_DIM + r] / Rtr[t * R_DIM + r];
    float lx = logf(x);
    acc += -lx - logf(sc) - 0.5f * LOG2PI_F - lx * lx / (2.0f * sc * sc);
  }
  sred[threadIdx.x] = acc; __syncthreads();
  for (int s = 128; s >= 1; s >>= 1) {
    if (threadIdx.x < s) sred[threadIdx.x] += sred[threadIdx.x + s];
    __syncthreads();
  }
  if (threadIdx.x == 0) atomicAdd(out, sred[0]);
}

// ---------------------------------------------------------------------------
// Case + death negbin: one block per (t, r); reduce inf_sum over S.
// ---------------------------------------------------------------------------
__global__ void casedeath_ll_k(const float* __restrict__ inf,
                               const float* __restrict__ crt,
                               const float* __restrict__ crr,
                               const float* __restrict__ case_data,
                               const float* __restrict__ death_data,
                               const float* __restrict__ case_od_p,
                               const float* __restrict__ death_od_p,
                               float* __restrict__ out) {
  __shared__ float sred[256];
  const int tr = blockIdx.x;                 // t*R + r
  const int t = tr / R_DIM, r = tr % R_DIM;
  float s = 0.0f;
  for (int i = threadIdx.x; i < S_DIM; i += 256)
    s += inf[(size_t)tr * S_DIM + i];
  sred[threadIdx.x] = s; __syncthreads();
  for (int k = 128; k >= 1; k >>= 1) {
    if (threadIdx.x < k) sred[threadIdx.x] += sred[threadIdx.x + k];
    __syncthreads();
  }
  if (threadIdx.x == 0) {
    float inf_sum = sred[0];
    float lp = negbin_lp(case_data[tr], inf_sum * crt[t] * crr[r], case_od_p[0])
             + negbin_lp(death_data[tr], inf_sum * DEATH_RATE, death_od_p[0]);
    atomicAdd(out, lp);
  }
}

// ---------------------------------------------------------------------------
// Strain multinomial: precompute fine->coarse assignment from the one-hot
// sample_matrix, then one block per (t, c).
// ---------------------------------------------------------------------------
__global__ void build_assign_k(const float* __restrict__ sm, int* __restrict__ assign) {
  int r = threadIdx.x;
  if (r >= R_DIM) return;
  int a = 0;
  for (int c = 0; c < RC_DIM; ++c)
    if (sm[c * R_DIM + r] > 0.5f) a = c;
  assign[r] = a;
}

__global__ void strain_ll_k(const float* __restrict__ inf,
                            const float* __restrict__ strain,
                            const int* __restrict__ assign,
                            float* __restrict__ out) {
  __shared__ int   sassign[R_DIM];
  __shared__ float r1[256], r2[256], r3[256];
  const int t = blockIdx.x, c = blockIdx.y;
  for (int i = threadIdx.x; i < R_DIM; i += 256) sassign[i] = assign[i];
  __syncthreads();

  float csum = 0.0f, nsum = 0.0f, part = 0.0f;
  for (int s = threadIdx.x; s < S_DIM; s += 256) {
    float cv = 1e-6f;
    for (int r = 0; r < R_DIM; ++r)
      if (sassign[r] == c) cv += inf[((size_t)t * R_DIM + r) * S_DIM + s];
    float k = strain[((size_t)t * RC_DIM + c) * S_DIM + s];
    csum += cv;
    nsum += k;
    part += k * logf(cv) - lgammaf(k + 1.0f);
  }
  r1[threadIdx.x] = csum; r2[threadIdx.x] = nsum; r3[threadIdx.x] = part;
  __syncthreads();
  for (int k = 128; k >= 1; k >>= 1) {
    if (threadIdx.x < k) {
      r1[threadIdx.x] += r1[threadIdx.x + k];
      r2[threadIdx.x] += r2[threadIdx.x + k];
      r3[threadIdx.x] += r3[threadIdx.x + k];
    }
    __syncthreads();
  }
  if (threadIdx.x == 0) {
    float total = lgammaf(r2[0] + 1.0f) + r3[0] - r2[0] * logf(r1[0]);
    atomicAdd(out, total);
  }
}

// ---------------------------------------------------------------------------
// launch
// ---------------------------------------------------------------------------
extern "C" void kernel_launch(void* const* d_in, const int* in_sizes, int n_in,
                              void* d_out, int out_size, void* d_ws, size_t ws_size,
                              hipStream_t stream) {
  const float* infections      = (const float*)d_in[0];
  const float* case_rate_time  = (const float*)d_in[1];
  const float* case_rate_region= (const float*)d_in[2];
  const float* R0              = (const float*)d_in[3];
  const float* Rs              = (const float*)d_in[4];
  const float* Rtr             = (const float*)d_in[5];
  const float* transit_rate    = (const float*)d_in[6];
  const float* mutation_rate   = (const float*)d_in[7];
  const float* infection_od    = (const float*)d_in[8];
  const float* case_od         = (const float*)d_in[9];
  const float* death_od        = (const float*)d_in[10];
  const float* drift_scale     = (const float*)d_in[11];
  const float* transit_data    = (const float*)d_in[12];
  const float* case_data       = (const float*)d_in[13];
  const float* death_data      = (const float*)d_in[14];
  const float* strain_data     = (const float*)d_in[15];
  const float* sample_matrix   = (const float*)d_in[16];
  const float* mutation_matrix = (const float*)d_in[17];
  float* out = (float*)d_out;

  // workspace carve-out (~124 MB)
  auto alignup = [](size_t x) { return (x + 255) & ~(size_t)255; };
  char* w = (char*)d_ws;
  size_t o = 0;
  float* tmpP     = (float*)(w + o); o = alignup(o + (size_t)(T_DIM - 1) * KP1 * NP * 2 * 4);
  float* pred1    = (float*)(w + o); o = alignup(o + (size_t)(T_DIM - 1) * MP * NP * 4);
  float* transitT = (float*)(w + o); o = alignup(o + (size_t)MP * R_DIM * 4);
  float* mutP     = (float*)(w + o); o = alignup(o + (size_t)KP2 * NP * 2 * 4);
  int*   assign   = (int*)  (w + o); o = alignup(o + (size_t)R_DIM * 4);

  init_out_k<<<1, 1, 0, stream>>>(out);

  build_transitT_k<<<(MP * R_DIM + 255) / 256, 256, 0, stream>>>(
      transit_data, transit_rate, transitT);
  build_mutP_k<<<(KP2 * NP + 255) / 256, 256, 0, stream>>>(
      mutation_matrix, mutation_rate, mutP);
  build_tmpP_k<<<((T_DIM - 1) * KP1 * NP + 255) / 256, 256, 0, stream>>>(
      infections, Rtr, Rs, R0, tmpP);

  // M: 19 tiles of 16; N: 8 strips of 64 (2 blocks x 4 waves)
  dim3 gg(MP / 16, 2, T_DIM - 1);
  gemm1_wmma_k<<<gg, 128, 0, stream>>>(transitT, tmpP, pred1);
  gemm2_wmma_ll_k<<<gg, 128, 0, stream>>>(pred1, mutP, infections, infection_od, out);

  drift_ll_k<<<117, 256, 0, stream>>>(Rtr, drift_scale, out);
  casedeath_ll_k<<<T_DIM * R_DIM, 256, 0, stream>>>(
      infections, case_rate_time, case_rate_region, case_data, death_data,
      case_od, death_od, out);

  build_assign_k<<<1, R_DIM, 0, stream>>>(sample_matrix, assign);
  dim3 gs(T_DIM, RC_DIM);
  strain_ll_k<<<gs, 256, 0, stream>>>(infections, strain_data, assign, out);
}